// UserEncoder_29377576304753
// MI455X (gfx1250) — compile-verified
//
#include <hip/hip_runtime.h>
#include <hip/hip_bf16.h>
#include <math.h>

typedef __attribute__((ext_vector_type(2))) float v2f;
typedef __attribute__((ext_vector_type(8))) float v8f;

#define BATCH   512
#define S_LEN   100
#define TOK_L   30
#define DMODEL  400
#define NHEAD   20
#define DHEAD   20
#define MHID    128
#define NEG_INF (-1.0e9f)

// ---------------------------------------------------------------------------
// 1) Embedding gather + mean pool over L=30 tokens -> click[B*S, 400]
// ---------------------------------------------------------------------------
__global__ __launch_bounds__(128) void embed_mean(const int* __restrict__ uc,
                                                  const float* __restrict__ emb,
                                                  float* __restrict__ click) {
  const int bs  = blockIdx.x;          // b*S + s
  const int tid = threadIdx.x;
  __shared__ int sidx[TOK_L];
  if (tid < TOK_L) sidx[tid] = uc[(size_t)bs * TOK_L + tid];
  __syncthreads();
  for (int col = tid; col < DMODEL; col += 128) {
    float acc = 0.f;
#pragma unroll 6
    for (int j = 0; j < TOK_L; ++j)
      acc += emb[(size_t)sidx[j] * DMODEL + col];
    click[(size_t)bs * DMODEL + col] = acc * (1.0f / (float)TOK_L);
  }
}

// ---------------------------------------------------------------------------
// 2) fp32 WMMA GEMM: C[M,N] = A[M,K] @ B[K,N]  (+bias, tanh optional)
//    One wave computes a 16 x (16*NT) macro-tile: the A fragment is loaded
//    once per K-step and reused for NT back-to-back independent WMMAs.
//    Requires M%16==0, N%(16*NT)==0, K%4==0.
// ---------------------------------------------------------------------------
template <int NT, int FUSE_TANH>
__global__ __launch_bounds__(32) void gemm16_wmma(const float* __restrict__ A,
                                                  const float* __restrict__ Bm,
                                                  float* __restrict__ C,
                                                  int M, int N, int K,
                                                  const float* __restrict__ bias) {
  const int lane = threadIdx.x;
  const int half = lane >> 4;
  const int l    = lane & 15;
  const int n0   = blockIdx.x * (16 * NT);
  const int m0   = blockIdx.y * 16;

  const float* arow = A + (size_t)(m0 + l) * K;   // A-frag row = lane&15
  v8f c[NT];
#pragma unroll
  for (int t = 0; t < NT; ++t) c[t] = (v8f){};

  for (int k = 0; k < K; k += 4) {
    const int ka = k + 2 * half;                  // half selects K pair
    v2f a; a.x = arow[ka]; a.y = arow[ka + 1];
    const float* b0 = Bm + (size_t)ka       * N + n0 + l;
    const float* b1 = Bm + (size_t)(ka + 1) * N + n0 + l;
#pragma unroll
    for (int t = 0; t < NT; ++t) {
      v2f b; b.x = b0[16 * t]; b.y = b1[16 * t];
      c[t] = __builtin_amdgcn_wmma_f32_16x16x4_f32(false, a, false, b,
                                                   (short)0, c[t], false, false);
    }
  }

#pragma unroll
  for (int t = 0; t < NT; ++t) {
    const int nc = n0 + 16 * t + l;
    const float bv = bias ? bias[nc] : 0.f;
#pragma unroll
    for (int i = 0; i < 8; ++i) {
      float v = c[t][i] + bv;
      if (FUSE_TANH) v = tanhf(v);
      C[(size_t)(m0 + i + 8 * half) * N + nc] = v;  // C-frag: VGPR i -> row i/i+8
    }
  }
}

// ---------------------------------------------------------------------------
// 3) Fused multi-head attention, flash-style online softmax, per (b, h).
//    Block = 7 waves x 32 lanes; wave w owns query rows [16w, 16w+16).
//    S padded 100 -> 112 (7 tiles), Dh=20 (5 K-steps of 4), PV Dh padded -> 32.
//    Tricks:
//      - addresses CLAMPED in-bounds; invalid data neutralized by the mask
//        (p = exp(x + NEG_INF) = 0) and by p=0 coefficients in the PV WMMA
//      - RAW logits go through the LDS transpose; after transpose lane l
//        (+lane l+16) holds row l, so the row max needs ONE shfl_xor(16);
//        exp is applied at A-fragment read
//      - per-row corr is redistributed to C-layout with 8 independent shfls
//      - padded V col 20 is a constant 1.0 "ones column": o1[:,20]
//        accumulates sum_k p inside the WMMA (flash rescale included)
// ---------------------------------------------------------------------------
__global__ __launch_bounds__(224) void attn_kernel(const float* __restrict__ Q,
                                                   const float* __restrict__ K,
                                                   const float* __restrict__ V,
                                                   const int* __restrict__ seq_len,
                                                   float* __restrict__ ctx) {
  const int h    = blockIdx.x;
  const int b    = blockIdx.y;
  const int wave = threadIdx.y;        // 0..6
  const int lane = threadIdx.x;
  const int half = lane >> 4;
  const int l    = lane & 15;
  const int slen = seq_len[b];

  __shared__ float p_s[7][16][17];     // per-wave logits-tile transpose stage

  const int    m0   = wave * 16;
  const size_t base = (size_t)b * S_LEN * DMODEL + (size_t)h * DHEAD;

  // Preload Q A-fragments (row clamped; padded output rows are never stored)
  v2f aq[5];
  {
    const int srow = m0 + l;
    const int src  = (srow < S_LEN) ? srow : (S_LEN - 1);
    const size_t qr = base + (size_t)src * DMODEL;
#pragma unroll
    for (int t = 0; t < 5; ++t) {
      const int ka = 4 * t + 2 * half;
      aq[t].x = Q[qr + ka];
      aq[t].y = Q[qr + ka + 1];
    }
  }

  float rmL = -3.0e38f;                // running max of row l (lane-indexed)
  v8f o0 = {}, o1 = {};

  const float scale = 0.22360679774997896f;      // 1/sqrt(20)

  // lane-constant helpers for the upper PV tile (cols 16..31)
  const int   cn    = 16 + l;
  const int   cnc   = (cn < DHEAD) ? cn : 0;        // clamped column
  const float colm  = (cn < DHEAD) ? 1.0f : 0.0f;   // keep real cols 16..19
  const float onesv = (cn == DHEAD) ? 1.0f : 0.0f;  // ones column at col 20

  for (int blk = 0; blk < 7; ++blk) {
    const int n0    = blk * 16;
    const int scol  = n0 + l;
    const int scolc = (scol < S_LEN) ? scol : (S_LEN - 1);   // clamped k row
    const size_t kr = base + (size_t)scolc * DMODEL;

    // ---- logits tile: S = Q(16x20) @ K^T(20x16)  (unconditional loads)
    v8f c = {};
#pragma unroll
    for (int t = 0; t < 5; ++t) {
      const int ka = 4 * t + 2 * half;
      v2f bk;
      bk.x = K[kr + ka];
      bk.y = K[kr + ka + 1];
      c = __builtin_amdgcn_wmma_f32_16x16x4_f32(false, aq[t], false, bk,
                                                (short)0, c, false, false);
    }

    // ---- store RAW masked logits through the transpose stage
    const float maskv = (scol < slen) ? 1.0f : NEG_INF;   // padded cols masked too
#pragma unroll
    for (int i = 0; i < 8; ++i)
      p_s[wave][i + 8 * half][l] = c[i] * scale + maskv;  // C-layout -> LDS
    asm volatile("s_wait_dscnt 0x0" ::: "memory");        // same-wave LDS RAW

    // ---- read back in A-fragment layout: lane l (+ lane l+16) = row l
    v2f raw[4];
#pragma unroll
    for (int t = 0; t < 4; ++t) {
      const int kk = 4 * t + 2 * half;
      raw[t].x = p_s[wave][l][kk];
      raw[t].y = p_s[wave][l][kk + 1];
    }

    // ---- row max: 7 local v_max + ONE cross-half shuffle (covers all rows)
    float rmax = fmaxf(fmaxf(fmaxf(raw[0].x, raw[0].y), fmaxf(raw[1].x, raw[1].y)),
                       fmaxf(fmaxf(raw[2].x, raw[2].y), fmaxf(raw[3].x, raw[3].y)));
    rmax = fmaxf(rmax, __shfl_xor(rmax, 16, 32));
    const float nm    = fmaxf(rmL, rmax);
    const float corrL = __expf(rmL - nm);      // 0 on first block (-3e38 - nm)
    rmL = nm;

    // ---- redistribute corr (lane-indexed rows) to C-layout accumulators
#pragma unroll
    for (int i = 0; i < 8; ++i) {
      const float corrC = __shfl(corrL, i + 8 * half, 32);  // row i + 8*half
      o0[i] *= corrC;
      o1[i] *= corrC;                           // also rescales the running sum
    }

    // ---- PV: O(16x32) += P(16x16) @ [Vblk | ones](16x32)  (K=16 -> 4 steps)
#pragma unroll
    for (int t = 0; t < 4; ++t) {
      const int kk  = 4 * t + 2 * half;
      v2f ap;                               // exp applied at A-fragment read
      ap.x = __expf(raw[t].x - nm);         // exactly 0 on masked/padded cols
      ap.y = __expf(raw[t].y - nm);
      const int sr0 = n0 + kk, sr1 = n0 + kk + 1;
      const int s0c = (sr0 < S_LEN) ? sr0 : (S_LEN - 1);   // clamped V rows:
      const int s1c = (sr1 < S_LEN) ? sr1 : (S_LEN - 1);   // their p coeffs are 0
      const float* vr0 = V + base + (size_t)s0c * DMODEL;
      const float* vr1 = V + base + (size_t)s1c * DMODEL;
      {
        v2f bv;                             // n-tile 0: ctx cols 0..15
        bv.x = vr0[l];
        bv.y = vr1[l];
        o0 = __builtin_amdgcn_wmma_f32_16x16x4_f32(false, ap, false, bv,
                                                   (short)0, o0, false, false);
      }
      {
        v2f bv;                             // n-tile 1: cols 16..19 real,
        bv.x = vr0[cnc] * colm + onesv;     //           col 20 = ones column
        bv.y = vr1[cnc] * colm + onesv;
        o1 = __builtin_amdgcn_wmma_f32_16x16x4_f32(false, ap, false, bv,
                                                   (short)0, o1, false, false);
      }
    }
  }

  // ---- normalize by the WMMA-accumulated row sum (lane 4 / lane 20) + store
  const int sumLane = (lane & 16) | 4;      // half0 -> lane 4, half1 -> lane 20
#pragma unroll
  for (int i = 0; i < 8; ++i) {
    const float rsum = __shfl(o1[i], sumLane, 32);
    const int srow = m0 + i + 8 * half;
    if (srow < S_LEN) {
      const float inv = 1.0f / rsum;        // >0: col 0 is always valid
      ctx[base + (size_t)srow * DMODEL + l] = o0[i] * inv;
      if (cn < DHEAD)
        ctx[base + (size_t)srow * DMODEL + cn] = o1[i] * inv;
    }
  }
}

// ---------------------------------------------------------------------------
// 4) fc2 + additive-attention pooling: out[b,:] = sum_s softmax_s(alpha) * ctx
// ---------------------------------------------------------------------------
__global__ __launch_bounds__(128) void pool_kernel(const float* __restrict__ e,
                                                   const float* __restrict__ fc2w,
                                                   const float* __restrict__ fc2b,
                                                   const float* __restrict__ ctx,
                                                   const int* __restrict__ seq_len,
                                                   float* __restrict__ out) {
  const int b   = blockIdx.x;
  const int tid = threadIdx.x;
  __shared__ float wsh[S_LEN];
  __shared__ float red[2];
  const int slen = seq_len[b];

  if (tid < S_LEN) {
    const float* ev = e + ((size_t)b * S_LEN + tid) * MHID;
    float acc = fc2b[0];
#pragma unroll 8
    for (int j = 0; j < MHID; ++j) acc += ev[j] * fc2w[j];
    acc += (tid < slen) ? 1.0f : NEG_INF;
    wsh[tid] = acc;
  }
  __syncthreads();
  if (tid == 0) {
    float mx = -3.0e38f;
    for (int s = 0; s < S_LEN; ++s) mx = fmaxf(mx, wsh[s]);
    red[0] = mx;
  }
  __syncthreads();
  const float mx = red[0];
  if (tid < S_LEN) wsh[tid] = __expf(wsh[tid] - mx);
  __syncthreads();
  if (tid == 0) {
    float sm = 0.f;
    for (int s = 0; s < S_LEN; ++s) sm += wsh[s];
    red[1] = 1.0f / sm;
  }
  __syncthreads();
  const float inv = red[1];
  for (int col = tid; col < DMODEL; col += 128) {
    float acc = 0.f;
    for (int s = 0; s < S_LEN; ++s)
      acc += wsh[s] * ctx[((size_t)b * S_LEN + s) * DMODEL + col];
    out[(size_t)b * DMODEL + col] = acc * inv;
  }
}

// ---------------------------------------------------------------------------
// Launch
// ---------------------------------------------------------------------------
extern "C" void kernel_launch(void* const* d_in, const int* in_sizes, int n_in,
                              void* d_out, int out_size, void* d_ws, size_t ws_size,
                              hipStream_t stream) {
  const int*   user_click = (const int*)d_in[0];
  const int*   seq_len    = (const int*)d_in[1];
  const float* emb        = (const float*)d_in[2];
  const float* wq         = (const float*)d_in[3];
  const float* wk         = (const float*)d_in[4];
  const float* wv         = (const float*)d_in[5];
  const float* fc1_w      = (const float*)d_in[6];
  const float* fc1_b      = (const float*)d_in[7];
  const float* fc2_w      = (const float*)d_in[8];
  const float* fc2_b      = (const float*)d_in[9];
  float*       out        = (float*)d_out;

  const size_t BS      = (size_t)BATCH * S_LEN;        // 51200
  const size_t szClick = BS * DMODEL * sizeof(float);  // 81.92 MB
  (void)ws_size; (void)in_sizes; (void)n_in; (void)out_size;

  char* ws     = (char*)d_ws;
  float* click = (float*)(ws);                   // later reused as ctx
  float* Qb    = (float*)(ws + 1 * szClick);
  float* Kb    = (float*)(ws + 2 * szClick);
  float* Vb    = (float*)(ws + 3 * szClick);
  float* Eb    = (float*)(ws + 4 * szClick);

  // 1) news encoder (gather + mean)
  embed_mean<<<(int)BS, 128, 0, stream>>>(user_click, emb, click);

  // 2) QKV projections: [51200,400] @ [400,400], 16x80 macro-tile per wave
  dim3 gQKV(DMODEL / (16 * 5), (int)(BS / 16));        // (5, 3200)
  gemm16_wmma<5, 0><<<gQKV, 32, 0, stream>>>(click, wq, Qb, (int)BS, DMODEL, DMODEL, nullptr);
  gemm16_wmma<5, 0><<<gQKV, 32, 0, stream>>>(click, wk, Kb, (int)BS, DMODEL, DMODEL, nullptr);
  gemm16_wmma<5, 0><<<gQKV, 32, 0, stream>>>(click, wv, Vb, (int)BS, DMODEL, DMODEL, nullptr);

  // 3) fused MHA (ctx overwrites click, which is dead after QKV)
  attn_kernel<<<dim3(NHEAD, BATCH), dim3(32, 7), 0, stream>>>(Qb, Kb, Vb, seq_len, click);

  // 4) fc1 + tanh: [51200,400] @ [400,128], 16x64 macro-tile per wave
  gemm16_wmma<4, 1><<<dim3(MHID / (16 * 4), (int)(BS / 16)), 32, 0, stream>>>(
      click, fc1_w, Eb, (int)BS, MHID, DMODEL, fc1_b);

  // 5) fc2 + masked softmax pooling -> out[512,400]
  pool_kernel<<<BATCH, 128, 0, stream>>>(Eb, fc2_w, fc2_b, click, seq_len, out);
}